// HE_22840636080958
// MI455X (gfx1250) — compile-verified
//
#include <hip/hip_runtime.h>

// Histogram equalization, 3 channels of 4096x4096 f32.
// Memory-bound: ~600 MB traffic -> ~26us at 23.3 TB/s HBM.
// CDNA5 path: 256-bin prefix sum done with V_WMMA_F32_16X16X4_F32 (exact:
// integer counts <= 2^24 are exact in f32, mask weights are 0/1).

typedef float v2f __attribute__((ext_vector_type(2)));
typedef float v8f __attribute__((ext_vector_type(8)));

#define NB 256

__device__ __forceinline__ int he_quant(float v) {
    int q = (int)floorf(v * 255.0f);
    q = q < 0 ? 0 : q;
    return q > 255 ? 255 : q;
}

__global__ void he_zero_hist(unsigned int* __restrict__ g_hist) {
    int t = blockIdx.x * blockDim.x + threadIdx.x;
    if (t < 3 * NB) g_hist[t] = 0u;
}

// Pass 1: per-wave privatized LDS histograms, one global atomic per bin/block.
__global__ void he_hist(const float* __restrict__ x,
                        unsigned int* __restrict__ g_hist, int npix) {
    __shared__ unsigned int sub[8][NB];
    const int tid = threadIdx.x;
    const int ch  = blockIdx.y;
    const int w   = tid >> 5;            // wave32 on gfx1250
#pragma unroll
    for (int i = 0; i < 8; ++i) sub[i][tid] = 0u;
    __syncthreads();

    const float4* __restrict__ src = (const float4*)(x + (size_t)ch * (size_t)npix);
    const int nvec = npix >> 2;
    for (int i = blockIdx.x * blockDim.x + tid; i < nvec; i += gridDim.x * blockDim.x) {
        float4 v = src[i];
        atomicAdd(&sub[w][he_quant(v.x)], 1u);
        atomicAdd(&sub[w][he_quant(v.y)], 1u);
        atomicAdd(&sub[w][he_quant(v.z)], 1u);
        atomicAdd(&sub[w][he_quant(v.w)], 1u);
    }
    __syncthreads();

    unsigned int s = 0;
#pragma unroll
    for (int i = 0; i < 8; ++i) s += sub[i][tid];
    if (s) atomicAdd(&g_hist[ch * NB + tid], s);
}

// Pass 2: per-channel LUT. 256-bin inclusive scan via WMMA:
//   H[s][c] = hist[16s+c];  P = H x U (U upper-tri ones)  -> within-segment prefix
//   cdf     = Lstrict x Sbcast + P (accumulator seeded with P)
// All 8 waves compute redundantly so EXEC is all-ones for every WMMA.
__global__ void he_lut(const unsigned int* __restrict__ g_hist,
                       float* __restrict__ g_lut, int npix) {
    __shared__ float histf[NB];
    __shared__ float segsum[16];
    __shared__ float cdf[NB];
    __shared__ int   i0_s;

    const int tid = threadIdx.x;
    const int ch  = blockIdx.x;

    histf[tid] = (float)g_hist[ch * NB + tid];
    if (tid == 0) i0_s = NB;
    __syncthreads();

    if (tid < 16) {
        float s = 0.0f;
#pragma unroll
        for (int c = 0; c < 16; ++c) s += histf[tid * 16 + c];
        segsum[tid] = s;
    }
    __syncthreads();

    // lane geometry (wave32): A rows / B-C-D cols live in lanes 0..15 & 16..31
    const int L  = tid & 31;
    const int m  = L & 15;               // A row index
    const int n  = L & 15;               // B/C/D column index
    const int kh = (L & 16) ? 2 : 0;     // K half selected by lane group
    const int segBase = (L & 16) ? 8 : 0;

    v8f acc = {};

    // Chain 1: P = H x U   (4 x 16x16x4)
#pragma unroll
    for (int kk = 0; kk < 4; ++kk) {
        const int k0 = 4 * kk + kh, k1 = k0 + 1;
        v2f a, b;
        a.x = histf[m * 16 + k0];
        a.y = histf[m * 16 + k1];
        b.x = (k0 <= n) ? 1.0f : 0.0f;
        b.y = (k1 <= n) ? 1.0f : 0.0f;
        acc = __builtin_amdgcn_wmma_f32_16x16x4_f32(false, a, false, b,
                                                    (short)0, acc, false, false);
    }
    // Chain 2: cdf = Lstrict x Sbcast + P   (4 x 16x16x4, C seeded with P)
#pragma unroll
    for (int kk = 0; kk < 4; ++kk) {
        const int k0 = 4 * kk + kh, k1 = k0 + 1;
        v2f a, b;
        a.x = (k0 < m) ? 1.0f : 0.0f;
        a.y = (k1 < m) ? 1.0f : 0.0f;
        b.x = segsum[k0];
        b.y = segsum[k1];
        acc = __builtin_amdgcn_wmma_f32_16x16x4_f32(false, a, false, b,
                                                    (short)0, acc, false, false);
    }

#pragma unroll
    for (int i = 0; i < 8; ++i)
        cdf[(segBase + i) * 16 + n] = acc[i];   // all waves write identical values
    __syncthreads();

    if (histf[tid] > 0.0f) atomicMin(&i0_s, tid);   // first populated bin
    __syncthreads();

    int i0 = i0_s;
    if (i0 >= NB) i0 = 0;
    const float cdf0  = cdf[i0];
    const float scale = 255.0f / ((float)npix - histf[i0]);
    float v = rintf((cdf[tid] - cdf0) * scale);     // RNE matches jnp.round
    v = v < 0.0f ? 0.0f : (v > 255.0f ? 255.0f : v);
    g_lut[ch * NB + tid] = v * (1.0f / 255.0f);
}

// Pass 3: remap through LDS-staged LUT, float4 in / float4 out.
__global__ void he_remap(const float* __restrict__ x,
                         const float* __restrict__ g_lut,
                         float* __restrict__ out, int npix) {
    __shared__ float lut[NB];
    const int tid = threadIdx.x;
    const int ch  = blockIdx.y;
    lut[tid] = g_lut[ch * NB + tid];
    __syncthreads();

    const float4* __restrict__ src = (const float4*)(x   + (size_t)ch * (size_t)npix);
    float4* __restrict__       dst = (float4*)      (out + (size_t)ch * (size_t)npix);
    const int nvec = npix >> 2;
    for (int i = blockIdx.x * blockDim.x + tid; i < nvec; i += gridDim.x * blockDim.x) {
        float4 v = src[i];
        float4 r;
        r.x = lut[he_quant(v.x)];
        r.y = lut[he_quant(v.y)];
        r.z = lut[he_quant(v.z)];
        r.w = lut[he_quant(v.w)];
        dst[i] = r;
    }
}

extern "C" void kernel_launch(void* const* d_in, const int* in_sizes, int n_in,
                              void* d_out, int out_size, void* d_ws, size_t ws_size,
                              hipStream_t stream) {
    const float* x  = (const float*)d_in[0];
    float* out      = (float*)d_out;
    const int total = in_sizes[0];       // 3 * 4096 * 4096
    const int npix  = total / 3;

    unsigned int* g_hist = (unsigned int*)d_ws;                 // 3*256 u32
    float*        g_lut  = (float*)((char*)d_ws + 4096);        // 3*256 f32

    he_zero_hist<<<dim3(3), dim3(256), 0, stream>>>(g_hist);

    dim3 hgrid(1024, 3);
    he_hist<<<hgrid, dim3(256), 0, stream>>>(x, g_hist, npix);

    he_lut<<<dim3(3), dim3(256), 0, stream>>>(g_hist, g_lut, npix);

    dim3 rgrid(2048, 3);
    he_remap<<<rgrid, dim3(256), 0, stream>>>(x, g_lut, out, npix);
}